// GroupRouter_70866960384583
// MI455X (gfx1250) — compile-verified
//
#include <hip/hip_runtime.h>
#include <hip/hip_bf16.h>

typedef __attribute__((ext_vector_type(2))) float v2f;
typedef __attribute__((ext_vector_type(8))) float v8f;
typedef unsigned int v4u __attribute__((ext_vector_type(4)));
typedef int v8i __attribute__((ext_vector_type(8)));
typedef int v4i __attribute__((ext_vector_type(4)));

#define D_MODEL 4096
#define N_EXP   64
#define TOKS    16      // tokens per workgroup (one WMMA M-tile)
#define THREADS 256     // 8 waves of 32
#define LN_EPS  1e-5f

// ---------------------------------------------------------------------------
// Prep: Wt[d][e] = gamma[d] * W[e][d]  (transposed, gamma folded)
//       c[e]    = b[e] + sum_d beta[d] * W[e][d]
//       accum[] = 0
// One block per expert.
// ---------------------------------------------------------------------------
__global__ __launch_bounds__(THREADS)
void router_prep(const float* __restrict__ W, const float* __restrict__ b,
                 const float* __restrict__ gamma, const float* __restrict__ beta,
                 float* __restrict__ Wt, float* __restrict__ cvec,
                 float* __restrict__ accum) {
    const int e   = blockIdx.x;
    const int tid = threadIdx.x;
    float part = 0.f;
    for (int d = tid; d < D_MODEL; d += THREADS) {
        float w = W[(size_t)e * D_MODEL + d];
        Wt[(size_t)d * N_EXP + e] = w * gamma[d];
        part += w * beta[d];
    }
    __shared__ float red[THREADS];
    red[tid] = part;
    __syncthreads();
    for (int s = THREADS / 2; s > 0; s >>= 1) {
        if (tid < s) red[tid] += red[tid + s];
        __syncthreads();
    }
    if (tid == 0) cvec[e] = red[0] + b[e];
    if (e == 0 && tid < N_EXP) accum[tid] = 0.f;
}

// ---------------------------------------------------------------------------
// Fused LayerNorm + gate GEMM (WMMA f32 16x16x4) + softmax + top-2 + scatter.
// One block = 16 tokens. The 256 KB x tile is DMA'd into LDS by the Tensor
// Data Mover (one tensor_load_to_lds per block), so x crosses HBM exactly
// once and never touches VGPRs on the way in.
// ---------------------------------------------------------------------------
__global__ __launch_bounds__(THREADS)
void router_main(const float* __restrict__ x, const float* __restrict__ Wt,
                 const float* __restrict__ cvec,
                 float* __restrict__ sparse_w, int* __restrict__ inds,
                 float* __restrict__ accum) {
    __shared__ float ldsX[TOKS * D_MODEL];   // 256 KB token tile
    __shared__ float meanArr[TOKS];
    __shared__ float rstdArr[TOKS];
    __shared__ float logits[TOKS * N_EXP];   // 4 KB
    __shared__ float cpart[4][8 * 32];       // upper-K-half partial C tiles

    const int tid  = threadIdx.x;
    const int wave = tid >> 5;
    const int lane = tid & 31;
    const size_t base = (size_t)blockIdx.x * TOKS * D_MODEL;

    // ---- Stage 1: TDM async tensor DMA of the x tile into LDS ----
    if (wave == 0) {
        const unsigned lds_base = (unsigned)(uintptr_t)(void*)ldsX; // LDS byte offset (low 32b of generic ptr)
        const unsigned long long ga = (unsigned long long)(uintptr_t)(x + base);
        // D# group 0: count=1 | lds_addr | global_addr[56:0] | type=2
        v4u g0;
        g0.x = 1u;
        g0.y = lds_base;
        g0.z = (unsigned)(ga & 0xFFFFFFFFu);
        g0.w = (unsigned)((ga >> 32) & 0x01FFFFFFu) | (2u << 30);
        // D# group 1: data_size=4B; tensor 4096x16 f32; tile 4096x16; stride 4096
        v8i g1;
        g1[0] = (int)(2u << 16);          // data_size = 2 (4 bytes)
        g1[1] = (int)(D_MODEL << 16);     // tensor_dim0[15:0] in bits [31:16]
        g1[2] = (int)(TOKS << 16);        // tensor_dim1[15:0] in bits [31:16]
        g1[3] = (int)(D_MODEL << 16);     // tile_dim0 in bits [31:16]
        g1[4] = TOKS;                     // tile_dim1
        g1[5] = D_MODEL;                  // tensor_dim0_stride[31:0]
        g1[6] = 0;
        g1[7] = 0;
        v4i gz = {0, 0, 0, 0};            // groups 2/3 unused (2-D tensor)
#if __has_include(<hip/amd_detail/amd_gfx1250_TDM.h>)
        v8i gz8 = {0, 0, 0, 0, 0, 0, 0, 0};
        __builtin_amdgcn_tensor_load_to_lds(g0, g1, gz, gz, gz8, 0);
#else
        __builtin_amdgcn_tensor_load_to_lds(g0, g1, gz, gz, 0);
#endif
        __builtin_amdgcn_s_wait_tensorcnt(0);
    }
    __syncthreads();   // releases the other 7 waves once the DMA landed

    // ---- Stage 2: per-token mean / rstd (wave handles 2 tokens) ----
    for (int t = wave * 2; t < wave * 2 + 2; ++t) {
        float s = 0.f, ss = 0.f;
        for (int k = lane; k < D_MODEL; k += 32) {
            float v = ldsX[t * D_MODEL + k];
            s += v; ss += v * v;
        }
        for (int off = 16; off > 0; off >>= 1) {
            s  += __shfl_xor(s,  off, 32);
            ss += __shfl_xor(ss, off, 32);
        }
        if (lane == 0) {
            float mu  = s / (float)D_MODEL;
            float var = ss / (float)D_MODEL - mu * mu;
            meanArr[t] = mu;
            rstdArr[t] = rsqrtf(var + LN_EPS);
        }
    }
    __syncthreads();

    // ---- Stage 3: WMMA GEMM, logits = xn @ Wt + c ----
    // wave -> (expert N-tile, K half). 512 x v_wmma_f32_16x16x4_f32 per wave.
    const int ntile = wave & 3;
    const int khalf = wave >> 2;
    const int mrow  = lane & 15;   // token row  (A: lanes 0-15 = M, 16-31 = M)
    const int hi    = lane >> 4;   // K sub-pair select (K+0,1 vs K+2,3)
    const int ncol  = ntile * 16 + (lane & 15);
    const float mu = meanArr[mrow];
    const float rs = rstdArr[mrow];

    v8f acc = {};
    const int k0   = khalf * (D_MODEL / 2);
    const int kend = k0 + (D_MODEL / 2);
    for (int k = k0; k < kend; k += 4) {
        const int ka = k + hi * 2;
        // A fragment: normalized x, VGPR0=K+0|K+2, VGPR1=K+1|K+3 (ISA layout)
        float2 xv = *(const float2*)&ldsX[mrow * D_MODEL + ka];
        v2f a;
        a.x = (xv.x - mu) * rs;
        a.y = (xv.y - mu) * rs;
        // B fragment: Wt rows ka, ka+1 for 16 consecutive experts (L2-hot)
        v2f bf;
        bf.x = Wt[(size_t)ka * N_EXP + ncol];
        bf.y = Wt[(size_t)(ka + 1) * N_EXP + ncol];
        acc = __builtin_amdgcn_wmma_f32_16x16x4_f32(
                  false, a, false, bf, (short)0, acc, false, false);
    }

    // combine the two K-halves through LDS
    if (khalf == 1) {
        #pragma unroll
        for (int v = 0; v < 8; ++v) cpart[ntile][v * 32 + lane] = acc[v];
    }
    __syncthreads();
    if (khalf == 0) {
        #pragma unroll
        for (int v = 0; v < 8; ++v) {
            // C/D layout: VGPR v, lanes 0-15 -> M=v, lanes 16-31 -> M=v+8
            int tok = v + 8 * hi;
            int e   = ntile * 16 + (lane & 15);
            logits[tok * N_EXP + e] = acc[v] + cpart[ntile][v * 32 + lane] + cvec[e];
        }
    }
    __syncthreads();

    // ---- Stage 4: softmax + top-2 + renormalized scatter (1 thread/token) ----
    if (tid < TOKS) {
        const int t = tid;
        const size_t gtok = (size_t)blockIdx.x * TOKS + t;
        float mx = -1e30f;
        for (int e = 0; e < N_EXP; ++e) mx = fmaxf(mx, logits[t * N_EXP + e]);
        float sum = 0.f;
        for (int e = 0; e < N_EXP; ++e) {
            float ex = __expf(logits[t * N_EXP + e] - mx);
            logits[t * N_EXP + e] = ex;
            sum += ex;
        }
        const float inv = 1.f / sum;
        float v1 = -1.f, v2 = -1.f; int i1 = 0, i2 = 0;
        for (int e = 0; e < N_EXP; ++e) {
            float p = logits[t * N_EXP + e] * inv;
            logits[t * N_EXP + e] = p;            // keep dense probs for lb loss
            if (p > v1)      { v2 = v1; i2 = i1; v1 = p; i1 = e; }
            else if (p > v2) { v2 = p;  i2 = e; }
        }
        const float den = v1 + v2 + 1e-8f;
        float* outrow = sparse_w + gtok * N_EXP;
        for (int e = 0; e < N_EXP; ++e) outrow[e] = 0.f;
        outrow[i1] = v1 / den;
        outrow[i2] = v2 / den;
        inds[gtok * 2]     = i1;
        inds[gtok * 2 + 1] = i2;
    }
    __syncthreads();

    // ---- per-expert partial sum of dense probs for the lb loss ----
    if (tid < N_EXP) {
        float s = 0.f;
        for (int t = 0; t < TOKS; ++t) s += logits[t * N_EXP + tid];
        atomicAdd(&accum[tid], s);
    }
}

// ---------------------------------------------------------------------------
// lb_loss = -sum_e mean_w[e] * log(mean_w[e] + 1e-8)
// ---------------------------------------------------------------------------
__global__ void router_lb(const float* __restrict__ accum, float* __restrict__ out_lb,
                          int ntok) {
    __shared__ float red[N_EXP];
    const int tid = threadIdx.x;
    float mw = accum[tid] / (float)ntok;
    red[tid] = -mw * __logf(mw + 1e-8f);
    __syncthreads();
    for (int s = N_EXP / 2; s > 0; s >>= 1) {
        if (tid < s) red[tid] += red[tid + s];
        __syncthreads();
    }
    if (tid == 0) *out_lb = red[0];
}

extern "C" void kernel_launch(void* const* d_in, const int* in_sizes, int n_in,
                              void* d_out, int out_size, void* d_ws, size_t ws_size,
                              hipStream_t stream) {
    const float* x     = (const float*)d_in[0];
    const float* gamma = (const float*)d_in[1];
    const float* beta  = (const float*)d_in[2];
    const float* W     = (const float*)d_in[3];
    const float* b     = (const float*)d_in[4];
    // d_in[5] = top_k (fixed at 2 by the reference's shapes)

    const int ntok = in_sizes[0] / D_MODEL;   // 16384

    // d_out layout: sparse_w [ntok*64] f32, indices [ntok*2] i32, lb_loss [1] f32
    float* sparse_w = (float*)d_out;
    int*   inds     = (int*)(sparse_w + (size_t)ntok * N_EXP);
    float* out_lb   = (float*)(sparse_w + (size_t)ntok * N_EXP + (size_t)ntok * 2);

    // workspace: Wt [4096*64], cvec [64], accum [64]
    float* Wt    = (float*)d_ws;
    float* cvec  = Wt + (size_t)D_MODEL * N_EXP;
    float* accum = cvec + N_EXP;

    router_prep<<<N_EXP, THREADS, 0, stream>>>(W, b, gamma, beta, Wt, cvec, accum);
    router_main<<<ntok / TOKS, THREADS, 0, stream>>>(x, Wt, cvec, sparse_w, inds, accum);
    router_lb<<<1, N_EXP, 0, stream>>>(accum, out_lb, ntok);
}